// GeometricCPU_45715631898883
// MI455X (gfx1250) — compile-verified
//
#include <hip/hip_runtime.h>
#include <math.h>
#include <stdint.h>

typedef float v2f __attribute__((ext_vector_type(2)));
typedef float v4f __attribute__((ext_vector_type(4)));
typedef float v8f __attribute__((ext_vector_type(8)));

#define WAVES_PER_BLOCK 8
#define TILES_PER_WAVE  8
#define LDS_STRIDE      20  // 16 + 4 pad: conflict-free, keeps 16B alignment

// ---- Cl(3,0,1) geometric-product sign: C[a,b,a^b] = gp_sign(a,b) ----
__device__ __forceinline__ float gp_sign(int a, int b) {
    if (a & b & 8) return 0.0f;          // e3*e3 = 0 (null direction)
    int s = 0, t = a >> 1;
    while (t) { s += __popc(t & b); t >>= 1; }
    return (s & 1) ? -1.0f : 1.0f;
}

// ---------------------------------------------------------------------------
// Setup: per row l compute combined sandwich matrix S[l] (16x16) and the
// softmax-blended color table (10x10). Tiny: 32 threads total.
// ---------------------------------------------------------------------------
__global__ void setup_kernel(const float* __restrict__ instruction,
                             const float* __restrict__ remap_tables,
                             const float* __restrict__ selector_w,
                             const float* __restrict__ selector_b,
                             float* __restrict__ Sws,   // [L][16][16]
                             float* __restrict__ Bws,   // [L][10][10]
                             int L) {
    int l = blockIdx.x * blockDim.x + threadIdx.x;
    if (l >= L) return;

    float instr[16];
    for (int i = 0; i < 16; ++i) instr[i] = instruction[l * 16 + i];

    // x = -0.5 * grade2(instruction)
    float x[16];
    for (int i = 0; i < 16; ++i)
        x[i] = (__popc(i) == 2) ? -0.5f * instr[i] : 0.0f;

    // M = exp(x), 16-term Taylor: term_{n} = gp(term_{n-1}, x)/n
    float term[16], acc[16];
    for (int i = 0; i < 16; ++i) { term[i] = (i == 0) ? 1.0f : 0.0f; acc[i] = term[i]; }
    for (int n = 1; n < 16; ++n) {
        float nt[16];
        for (int k = 0; k < 16; ++k) nt[k] = 0.0f;
        for (int a = 0; a < 16; ++a) {
            float ta = term[a];
            for (int b = 0; b < 16; ++b)
                nt[a ^ b] += gp_sign(a, b) * ta * x[b];
        }
        float inv = 1.0f / (float)n;
        for (int k = 0; k < 16; ++k) { term[k] = nt[k] * inv; acc[k] += term[k]; }
    }
    float M[16], Mr[16];
    for (int i = 0; i < 16; ++i) {
        M[i] = acc[i];
        int g = __popc(i);
        Mr[i] = (((g * (g - 1) / 2) & 1) ? -M[i] : M[i]);   // reverse sign
    }

    // S[j][k] = sum_i (M[j^i]*sign(j^i,j)) * (Mr[i^k]*sign(i,i^k))
    for (int j = 0; j < 16; ++j)
        for (int k = 0; k < 16; ++k) {
            float s = 0.0f;
            for (int i = 0; i < 16; ++i) {
                float lm = M[j ^ i]  * gp_sign(j ^ i, j);
                float rm = Mr[i ^ k] * gp_sign(i, i ^ k);
                s += lm * rm;
            }
            Sws[l * 256 + j * 16 + k] = s;
        }

    // selector softmax (K=4) -> blended 10x10 table
    float s0 = instr[0], s1 = instr[15];
    float logit[4], mx = -1e30f;
    for (int k = 0; k < 4; ++k) {
        logit[k] = s0 * selector_w[k * 2 + 0] + s1 * selector_w[k * 2 + 1] + selector_b[k];
        mx = fmaxf(mx, logit[k]);
    }
    float w[4], sum = 0.0f;
    for (int k = 0; k < 4; ++k) { w[k] = __expf(logit[k] - mx); sum += w[k]; }
    float inv = 1.0f / sum;
    for (int k = 0; k < 4; ++k) w[k] *= inv;
    for (int ij = 0; ij < 100; ++ij) {
        float b = 0.0f;
        for (int k = 0; k < 4; ++k) b += w[k] * remap_tables[k * 100 + ij];
        Bws[l * 100 + ij] = b;
    }
}

// ---------------------------------------------------------------------------
// Main kernel: grid-stride over 16-row tiles, double-buffered via
// GLOBAL_LOAD_ASYNC_TO_LDS_B128 (ASYNCcnt), WMMA f32 16x16x4 sandwich,
// then per-row color unit out of the LDS tile.
// ---------------------------------------------------------------------------
__global__ void __launch_bounds__(256)
sandwich_color_kernel(const float* __restrict__ state,
                      const float* __restrict__ Sws,
                      const float* __restrict__ Bws,
                      float* __restrict__ out,
                      int N, int numTiles) {
    __shared__ float lds[WAVES_PER_BLOCK][2][16 * LDS_STRIDE];
    __shared__ float bl[112];                       // 100 used: blended table

    const int l    = blockIdx.y;
    const int tid  = threadIdx.x;
    const int wave = tid >> 5;
    const int lane = tid & 31;

    // stage blended color table for this l (block-wide, read via LDS broadcast)
    if (tid < 100) bl[tid] = Bws[l * 100 + tid];
    __syncthreads();

    int t = blockIdx.x * WAVES_PER_BLOCK + wave;
    if (t >= numTiles) return;                      // wave-uniform
    const int tstride = gridDim.x * WAVES_PER_BLOCK;

    const int col   = lane & 15;
    const int khalf = (lane >> 4) << 1;             // 0 (lanes 0-15) or 2 (lanes 16-31)

    // ---- B fragments: S[l] per ISA 4x16 B layout, held for all tiles ----
    v2f bfrag[4];
    const float* Sl = Sws + l * 256;
#pragma unroll
    for (int kc = 0; kc < 4; ++kc) {
        int kb = kc * 4 + khalf;
        bfrag[kc].x = Sl[(kb + 0) * 16 + col];
        bfrag[kc].y = Sl[(kb + 1) * 16 + col];
    }

    const float* base = state + (size_t)l * N * 16;
    const int r  = lane >> 1;                       // staging row 0..15
    const int cb = (lane & 1) * 8;                  // staging comp base 0 or 8

    // per-lane LDS byte addresses for both buffers (generic-pointer low 32 bits
    // == LDS offset per the gfx1250 aperture layout)
    uint32_t ldsoff[2];
    ldsoff[0] = (uint32_t)(uintptr_t)&lds[wave][0][r * LDS_STRIDE + cb];
    ldsoff[1] = (uint32_t)(uintptr_t)&lds[wave][1][r * LDS_STRIDE + cb];

    // async-issue one 16x16 f32 tile into LDS buffer b (2 x b128 per lane,
    // INST_OFFSET applies to both the LDS and global addresses)
    auto issue = [&](int tile, int b) {
        int n0 = tile * 16;
        long ofs = ((n0 + r) < N) ? ((long)n0 * 16 + r * 16 + cb)
                                  : ((long)(N - 1) * 16 + cb);   // tail clamp
        const float* g = base + ofs;
        asm volatile(
            "global_load_async_to_lds_b128 %0, %1, off\n\t"
            "global_load_async_to_lds_b128 %0, %1, off offset:16"
            :: "v"(ldsoff[b]), "v"(g) : "memory");
    };

    int buf = 0;
    issue(t, buf);                                  // prime the pipeline

    for (; t < numTiles; t += tstride, buf ^= 1) {
        const int tn = t + tstride;
        const bool havenext = tn < numTiles;        // wave-uniform
        if (havenext) issue(tn, buf ^ 1);

        // async loads complete in order: <=2 outstanding leaves only the
        // prefetched tile in flight; the current tile's LDS writes are done.
        if (havenext) asm volatile("s_wait_asynccnt 0x2" ::: "memory");
        else          asm volatile("s_wait_asynccnt 0x0" ::: "memory");

        float* myLds = lds[wave][buf];
        const int n0 = t * 16;

        // ---- A fragments from LDS + 4x WMMA f32 16x16x4 (accumulate over K) ----
        v8f acc = {};
        {
            const int arow = lane & 15;
#pragma unroll
            for (int kc = 0; kc < 4; ++kc) {
                int c = kc * 4 + khalf;             // even -> 8B-aligned b64 read
                v2f a = *(const v2f*)(myLds + arow * LDS_STRIDE + c);
                acc = __builtin_amdgcn_wmma_f32_16x16x4_f32(
                    false, a, false, bfrag[kc], (short)0, acc, false, false);
            }
        }

        // ---- D (16x16 f32, ISA C/D layout) -> LDS tile ----
        {
            const int rbase = (lane >> 4) << 3;     // 0 or 8
#pragma unroll
            for (int rr = 0; rr < 8; ++rr)
                myLds[(rbase + rr) * LDS_STRIDE + col] = acc[rr];
        }

        // ---- color unit: one row per lane (lanes 0-15), reconverges after ----
        if (lane < 16) {
            const int row = lane;
            float c = myLds[row * LDS_STRIDE + 0] * 9.0f;
            float e[10], mx = -1e30f;
#pragma unroll
            for (int i = 0; i < 10; ++i) {
                float d = c - (float)i;
                e[i] = -4.0f * d * d;
                mx = fmaxf(mx, e[i]);
            }
            float sum = 0.0f;
#pragma unroll
            for (int i = 0; i < 10; ++i) { e[i] = __expf(e[i] - mx); sum += e[i]; }
            float inv = 1.0f / sum;
            float rem0 = 0.0f, dotc = 0.0f;
#pragma unroll
            for (int j = 0; j < 10; ++j) {
                float rj = 0.0f;
#pragma unroll
                for (int i = 0; i < 10; ++i) rj += e[i] * bl[i * 10 + j];
                rj *= inv;
                if (j == 0) rem0 = rj;
                dotc += rj * (float)j;
            }
            myLds[row * LDS_STRIDE + 0]  = dotc * (1.0f / 9.0f);
            myLds[row * LDS_STRIDE + 15] = 1.0f - rem0;
        }
        // same-wave LDS ops are program-ordered: no barrier needed

        // ---- coalesced b128 store ----
        {
            const int rg = n0 + r;
            if (rg < N) {
                float* dst = out + ((size_t)l * N + rg) * 16 + cb;
                *(v4f*)(dst)     = *(const v4f*)(myLds + r * LDS_STRIDE + cb);
                *(v4f*)(dst + 4) = *(const v4f*)(myLds + r * LDS_STRIDE + cb + 4);
            }
        }
    }
}

// ---------------------------------------------------------------------------
extern "C" void kernel_launch(void* const* d_in, const int* in_sizes, int n_in,
                              void* d_out, int out_size, void* d_ws, size_t ws_size,
                              hipStream_t stream) {
    const float* state       = (const float*)d_in[0];   // [L,N,16]
    const float* instruction = (const float*)d_in[1];   // [L,16]
    const float* remap       = (const float*)d_in[2];   // [K,10,10]
    const float* selw        = (const float*)d_in[3];   // [K,2]
    const float* selb        = (const float*)d_in[4];   // [K]
    float* out = (float*)d_out;

    const int L = in_sizes[1] / 16;                     // 32
    const int N = in_sizes[0] / (L * 16);               // 100000

    float* Sws = (float*)d_ws;                          // L*256 floats
    float* Bws = Sws + (size_t)L * 256;                 // L*100 floats

    setup_kernel<<<(L + 31) / 32, 32, 0, stream>>>(instruction, remap, selw, selb,
                                                   Sws, Bws, L);

    const int numTiles = (N + 15) / 16;
    const int tilesPerBlock = WAVES_PER_BLOCK * TILES_PER_WAVE;
    int gx = (numTiles + tilesPerBlock - 1) / tilesPerBlock;
    if (gx < 1) gx = 1;
    dim3 grid(gx, L);
    sandwich_color_kernel<<<grid, 256, 0, stream>>>(state, Sws, Bws, out, N, numTiles);
}